// GraphDecoder_32392643346499
// MI455X (gfx1250) — compile-verified
//
#include <hip/hip_runtime.h>

typedef __attribute__((ext_vector_type(2))) float v2f;
typedef __attribute__((ext_vector_type(8))) float v8f;

#define NN 20000
#define SS 60
#define HH 128
#define OO 128
#define EE 320000
#define SH (SS * HH)   // 7680 floats per node in x
#define LSTR 132       // padded LDS row stride (floats); 8B-aligned rows, conflict-free

// ---------------- Kernel 1: zero the aggregation workspace ----------------
__global__ void gd_zero_agg(float4* __restrict__ agg4, int n4) {
    int i = blockIdx.x * blockDim.x + threadIdx.x;
    if (i < n4) agg4[i] = make_float4(0.f, 0.f, 0.f, 0.f);
}

// ---------------- Kernel 2: edge scatter (segment_sum on s==0 slice) ------
// 32 lanes per edge; each lane moves 4 contiguous floats.
__global__ void gd_edge_scatter(const float* __restrict__ x,
                                const int* __restrict__ ei,
                                float* __restrict__ agg) {
    long long idx = (long long)blockIdx.x * blockDim.x + threadIdx.x;
    if (idx >= (long long)EE * 32) return;
    int e = (int)(idx >> 5);
    int q = (int)(idx & 31);
    long long src = (long long)ei[e];
    long long dst = (long long)ei[EE + e];
    const float4 v = *(const float4*)(x + src * SH + q * 4);
    float* a = agg + dst * HH + q * 4;
    atomicAdd(a + 0, v.x);
    atomicAdd(a + 1, v.y);
    atomicAdd(a + 2, v.z);
    atomicAdd(a + 3, v.w);
}

// ---------------- Kernel 3: per-node 2-layer MLP via WMMA fp32 ------------
// Block = 64 threads = 2 waves; each wave handles 16 nodes.
// Loop order: kk outer, 8 accumulators live -> each A frag loaded once
// (single ds_load_b64); all weight loads are immediate-offset from one base.
__global__ __launch_bounds__(64)
void gd_node_mlp(const float* __restrict__ x, const float* __restrict__ agg,
                 const float* __restrict__ W1, const float* __restrict__ b1,
                 const float* __restrict__ W2, const float* __restrict__ b2,
                 float* __restrict__ out) {
    __shared__ float hA[2][16 * LSTR];   // input h tiles (row-major, padded)
    __shared__ float hB[2][16 * LSTR];   // post-layer-1 tiles

    const int wave = threadIdx.x >> 5;
    const int lane = threadIdx.x & 31;
    const int m    = lane & 15;          // row (A) / column (B,C) index
    const int half = lane >> 4;          // lane group 0/1
    const int base = blockIdx.x * 32 + wave * 16;

    // Stage 1: h = x[:,0,:] + agg  -> LDS (16 rows x 128 cols per wave)
    for (int r = 0; r < 16; ++r) {
        int node = base + r;
        float4 v = make_float4(0.f, 0.f, 0.f, 0.f);
        if (node < NN) {
            const float4 xv = *(const float4*)(x + (long long)node * SH + lane * 4);
            const float4 av = *(const float4*)(agg + (long long)node * HH + lane * 4);
            v.x = xv.x + av.x; v.y = xv.y + av.y;
            v.z = xv.z + av.z; v.w = xv.w + av.w;
        }
        float* p = &hA[wave][r * LSTR + lane * 4];
        p[0] = v.x; p[1] = v.y; p[2] = v.z; p[3] = v.w;
    }
    __syncthreads();

    // ---- Stage 2: h1 = relu(h @ W1 + b1) -> LDS ----
    {
        // Per-lane constant bases: A frags at hA[wave][m*LSTR + half*2 + kk*4],
        // B frags at W1[half*2*OO + m + kk*4*OO + t*16] (all-immediate offsets).
        const float* __restrict__ ha = &hA[wave][m * LSTR + half * 2];
        const float* __restrict__ wb = W1 + half * 2 * OO + m;
        v8f acc[8];
        #pragma unroll
        for (int t = 0; t < 8; ++t) acc[t] = (v8f){};
        #pragma unroll
        for (int kk = 0; kk < 32; ++kk) {
            const v2f a = *(const v2f*)(ha + kk * 4);
            #pragma unroll
            for (int t = 0; t < 8; ++t) {
                v2f b;
                b.x = wb[kk * 4 * OO + t * 16];
                b.y = wb[(kk * 4 + 1) * OO + t * 16];
                acc[t] = __builtin_amdgcn_wmma_f32_16x16x4_f32(
                             false, a, false, b, (short)0, acc[t], false, false);
            }
        }
        #pragma unroll
        for (int t = 0; t < 8; ++t) {
            const int n = t * 16 + m;
            const float bias = b1[n];
            #pragma unroll
            for (int r = 0; r < 8; ++r) {   // C layout: VGPR r -> row r + 8*half
                float val = acc[t][r] + bias;
                hB[wave][(r + half * 8) * LSTR + n] = val > 0.f ? val : 0.f;
            }
        }
    }
    __syncthreads();

    // ---- Stage 3: out = relu(h1 @ W2 + b2) -> global ----
    {
        const float* __restrict__ ha = &hB[wave][m * LSTR + half * 2];
        const float* __restrict__ wb = W2 + half * 2 * OO + m;
        v8f acc[8];
        #pragma unroll
        for (int t = 0; t < 8; ++t) acc[t] = (v8f){};
        #pragma unroll
        for (int kk = 0; kk < 32; ++kk) {
            const v2f a = *(const v2f*)(ha + kk * 4);
            #pragma unroll
            for (int t = 0; t < 8; ++t) {
                v2f b;
                b.x = wb[kk * 4 * OO + t * 16];
                b.y = wb[(kk * 4 + 1) * OO + t * 16];
                acc[t] = __builtin_amdgcn_wmma_f32_16x16x4_f32(
                             false, a, false, b, (short)0, acc[t], false, false);
            }
        }
        #pragma unroll
        for (int t = 0; t < 8; ++t) {
            const int n = t * 16 + m;
            const float bias = b2[n];
            #pragma unroll
            for (int r = 0; r < 8; ++r) {
                int node = base + r + half * 8;
                if (node < NN) {
                    float val = acc[t][r] + bias;
                    out[(long long)node * OO + n] = val > 0.f ? val : 0.f;
                }
            }
        }
    }
}

extern "C" void kernel_launch(void* const* d_in, const int* in_sizes, int n_in,
                              void* d_out, int out_size, void* d_ws, size_t ws_size,
                              hipStream_t stream) {
    const float* x  = (const float*)d_in[0];
    const int*   ei = (const int*)d_in[1];
    const float* W1 = (const float*)d_in[2];
    const float* b1 = (const float*)d_in[3];
    const float* W2 = (const float*)d_in[4];
    const float* b2 = (const float*)d_in[5];
    float* out = (float*)d_out;
    float* agg = (float*)d_ws;           // NN*HH floats = 10.24 MB

    (void)in_sizes; (void)n_in; (void)out_size; (void)ws_size;

    // 1) zero agg
    {
        int n4 = (NN * HH) / 4;          // 640,000 float4s
        int blocks = (n4 + 255) / 256;
        gd_zero_agg<<<blocks, 256, 0, stream>>>((float4*)agg, n4);
    }
    // 2) edge scatter
    {
        long long total = (long long)EE * 32;   // 10.24M lanes
        int blocks = (int)((total + 255) / 256);
        gd_edge_scatter<<<blocks, 256, 0, stream>>>(x, ei, agg);
    }
    // 3) node MLP (2 waves/block, 32 nodes/block)
    {
        int blocks = (NN + 31) / 32;     // 625
        gd_node_mlp<<<blocks, 64, 0, stream>>>(x, agg, W1, b1, W2, b2, out);
    }
}